// TransductiveParameter_5274219840156
// MI455X (gfx1250) — compile-verified
//
#include <hip/hip_runtime.h>

typedef float v2f __attribute__((ext_vector_type(2)));
typedef float v4f __attribute__((ext_vector_type(4)));
typedef float v8f __attribute__((ext_vector_type(8)));

#define NATM 384
#define INF  64
#define OUTF 4
#define HID  32
#define RBF  16
#define HPC  1024   // 2*R*HID
#define OUTC 128    // 2*OUT_F*R

#define START_F 0.006737946999085467f                    // exp(-5)
#define DMEAN_F ((1.0f - START_F) / 15.0f)
#define BETA_F  (1.0f / ((0.125f*(1.0f-START_F))*(0.125f*(1.0f-START_F))))
#define PI_F    3.14159265358979323846f

// ---------------------------------------------------------------------------
// Kernel A: hp[m,j] = sum_k h[m,k] * W_fc[j,k]   (M=384, N=1024, K=64)
// One wave per 16x16 tile, V_WMMA_F32_16X16X4_F32, K stepped by 4.
// ---------------------------------------------------------------------------
__global__ __launch_bounds__(32) void hp_gemm(const float* __restrict__ h,
                                              const float* __restrict__ Wfc,
                                              float* __restrict__ hp) {
  const int t   = threadIdx.x;       // 0..31
  const int l16 = t & 15;
  const int hi  = t >> 4;            // 0 or 1
  const int m0  = blockIdx.y * 16;
  const int j0  = blockIdx.x * 16;

  v8f acc = {};
  for (int k0 = 0; k0 < INF; k0 += 4) {
    const int ka = k0 + hi * 2;
    v2f a, b;
    // A 16x4 layout: lane = M, VGPR0 holds K=2*hi, VGPR1 holds K=2*hi+1
    a.x = h[(m0 + l16) * INF + ka];
    a.y = h[(m0 + l16) * INF + ka + 1];
    // B 4x16 = W_fc^T tile: B[k, n] = Wfc[j0+n, k]
    b.x = Wfc[(j0 + l16) * INF + ka];
    b.y = Wfc[(j0 + l16) * INF + ka + 1];
    acc = __builtin_amdgcn_wmma_f32_16x16x4_f32(false, a, false, b,
                                                (short)0, acc, false, false);
  }
  // C/D layout: VGPR v -> row v + 8*hi, col = l16
#pragma unroll
  for (int v = 0; v < 8; ++v)
    hp[(m0 + v + hi * 8) * HPC + j0 + l16] = acc[v];
}

// ---------------------------------------------------------------------------
// Kernel B: fused pairwise reduction. One block per m.
//   S[r][b]  = sum_n smear[m,n,r]*dxn[m,n,b]
//   T[b][c]  = sum_n dxn[m,n,b] * (sum_r smear[m,n,r]*hp[n, col(r,c)])
//   bk/bq -> att[m, 0..31]
// c in [0,64): c<32 -> K columns (hp col r*32+c), else Q (512 + r*32 + c-32)
// ---------------------------------------------------------------------------
#define NB 64
__global__ __launch_bounds__(256) void pair_reduce(const float* __restrict__ x,
                                                   const float* __restrict__ hp,
                                                   float* __restrict__ att) {
  __shared__ float sm[NB][RBF];     // smear for current n-tile
  __shared__ float dn[NB][3];       // dxn  for current n-tile
  __shared__ float Tred[4][3][64];
  __shared__ float Sred[RBF][3];

  const int m   = blockIdx.x;
  const int tid = threadIdx.x;
  const float xm0 = x[m * 3 + 0];
  const float xm1 = x[m * 3 + 1];
  const float xm2 = x[m * 3 + 2];

  const int s_r = tid / 3, s_b = tid - 3 * s_r;   // valid for tid<48
  float s_acc = 0.0f;

  const int c    = tid & 63;
  const int sub  = tid >> 6;                      // 0..3 (n sub-range)
  const int cbase = (c < 32) ? c : (512 + (c - 32));
  float t0 = 0.0f, t1 = 0.0f, t2 = 0.0f;

  for (int n0 = 0; n0 < NATM; n0 += NB) {
    __syncthreads();
    // ---- phase 1: fill smear/dxn tile (4 threads per n, one r-group each)
    {
      const int nl = tid & 63;
      const int n  = n0 + nl;
      const int g  = tid >> 6;                    // r-group 0..3
      const float dx0 = xm0 - x[n * 3 + 0];
      const float dx1 = xm1 - x[n * 3 + 1];
      const float dx2 = xm2 - x[n * 3 + 2];
      const float nsq = dx0 * dx0 + dx1 * dx1 + dx2 * dx2 + 1e-5f;
      const float nrm = sqrtf(nsq);
      const float cut = (nrm < 5.0f)
                          ? 0.5f * (__cosf(nrm * (PI_F / 5.0f)) + 1.0f)
                          : 0.0f;
      const float e = __expf(-nrm);               // alpha = 1, CUT_LO = 0
      if (g == 0) {
        const float inv = 1.0f / nsq;             // dxn = dx / nsq
        dn[nl][0] = dx0 * inv;
        dn[nl][1] = dx1 * inv;
        dn[nl][2] = dx2 * inv;
      }
#pragma unroll
      for (int rr = 0; rr < 4; ++rr) {
        const int r = g * 4 + rr;
        const float d = e - (START_F + (float)r * DMEAN_F);
        sm[nl][r] = cut * __expf(-BETA_F * d * d);
      }
    }
    __syncthreads();
    // ---- S partials
    if (tid < 48) {
#pragma unroll 8
      for (int n = 0; n < NB; ++n) s_acc += sm[n][s_r] * dn[n][s_b];
    }
    // ---- T partials: each thread owns column c, 16 n's
    for (int ni = sub * 16; ni < sub * 16 + 16; ++ni) {
      const float* hr = hp + (size_t)(n0 + ni) * HPC + cbase;
      float P = 0.0f;
#pragma unroll
      for (int r = 0; r < RBF; ++r) P += sm[ni][r] * hr[r * 32];
      t0 += dn[ni][0] * P;
      t1 += dn[ni][1] * P;
      t2 += dn[ni][2] * P;
    }
  }

  Tred[sub][0][c] = t0;
  Tred[sub][1][c] = t1;
  Tred[sub][2][c] = t2;
  if (tid < 48) Sred[s_r][s_b] = s_acc;
  __syncthreads();

  if (tid < 32) {
    const int cc = tid;
    float bk[3], bq[3];
#pragma unroll
    for (int b = 0; b < 3; ++b) {
      bk[b] = Tred[0][b][cc] + Tred[1][b][cc] + Tred[2][b][cc] + Tred[3][b][cc];
      bq[b] = Tred[0][b][cc + 32] + Tred[1][b][cc + 32] +
              Tred[2][b][cc + 32] + Tred[3][b][cc + 32];
    }
    const float* hm = hp + (size_t)m * HPC;
#pragma unroll
    for (int r = 0; r < RBF; ++r) {
      const float hk = hm[r * 32 + cc];
      const float hq = hm[512 + r * 32 + cc];
#pragma unroll
      for (int b = 0; b < 3; ++b) {
        bk[b] += Sred[r][b] * hk;
        bq[b] += Sred[r][b] * hq;
      }
    }
    att[m * HID + cc] = bk[0] * bq[0] + bk[1] * bq[1] + bk[2] * bq[2];
  }
}

// ---------------------------------------------------------------------------
// Kernel C: out[m,j] = silu(att[m,:]) @ W_sum^T + b_sum   (384x128x32), WMMA
// ---------------------------------------------------------------------------
__global__ __launch_bounds__(32) void out_gemm(const float* __restrict__ att,
                                               const float* __restrict__ Wsum,
                                               const float* __restrict__ bsum,
                                               float* __restrict__ out) {
  const int t   = threadIdx.x;
  const int l16 = t & 15;
  const int hi  = t >> 4;
  const int m0  = blockIdx.y * 16;
  const int j0  = blockIdx.x * 16;

  v8f acc;
  const float bs = bsum[j0 + l16];   // bias: same value for every row of column
#pragma unroll
  for (int v = 0; v < 8; ++v) acc[v] = bs;

  for (int k0 = 0; k0 < HID; k0 += 4) {
    const int ka = k0 + hi * 2;
    const float a0 = att[(m0 + l16) * HID + ka];
    const float a1 = att[(m0 + l16) * HID + ka + 1];
    v2f a, b;
    a.x = a0 / (1.0f + __expf(-a0));   // silu
    a.y = a1 / (1.0f + __expf(-a1));
    b.x = Wsum[(j0 + l16) * HID + ka];
    b.y = Wsum[(j0 + l16) * HID + ka + 1];
    acc = __builtin_amdgcn_wmma_f32_16x16x4_f32(false, a, false, b,
                                                (short)0, acc, false, false);
  }
#pragma unroll
  for (int v = 0; v < 8; ++v)
    out[(m0 + v + hi * 8) * OUTC + j0 + l16] = acc[v];
}

// ---------------------------------------------------------------------------
// Kernel D: K2/Q2 emit — streaming non-temporal 128-bit stores (75.5 MB,
// HBM-bound; NT keeps L2 clean for hp/out which are the reuse-heavy data).
//   hh[m,r,o] = out[m, r*8+o]; K2[m,n,r,o]=hh[m,r,o]+hh[n,r,o] (o<4), Q2 o+4
// ---------------------------------------------------------------------------
__global__ __launch_bounds__(256) void emit(const float* __restrict__ out,
                                            float* __restrict__ O) {
  const unsigned total = (unsigned)NATM * NATM * RBF;   // 2359296
  unsigned i = blockIdx.x * 256u + threadIdx.x;
  if (i >= total) return;
  const unsigned r  = i & 15u;
  const unsigned nm = i >> 4;
  const unsigned n  = nm % NATM;
  const unsigned m  = nm / NATM;

  const v4f* o4 = (const v4f*)out;   // out as [384][32] float4
  const v4f mk = o4[m * 32 + r * 2];
  const v4f mq = o4[m * 32 + r * 2 + 1];
  const v4f nk = o4[n * 32 + r * 2];
  const v4f nq = o4[n * 32 + r * 2 + 1];

  const v4f k2 = mk + nk;
  const v4f q2 = mq + nq;

  v4f* O4 = (v4f*)O;
  __builtin_nontemporal_store(k2, O4 + i);           // K2 block
  __builtin_nontemporal_store(q2, O4 + total + i);   // Q2 block
}

// ---------------------------------------------------------------------------
extern "C" void kernel_launch(void* const* d_in, const int* in_sizes, int n_in,
                              void* d_out, int out_size, void* d_ws, size_t ws_size,
                              hipStream_t stream) {
  const float* x    = (const float*)d_in[0];
  const float* h    = (const float*)d_in[1];
  const float* Wfc  = (const float*)d_in[2];
  const float* Wsum = (const float*)d_in[3];
  const float* bsum = (const float*)d_in[4];

  float* hp  = (float*)d_ws;                 // 384*1024
  float* att = hp + (size_t)NATM * HPC;      // 384*32
  float* out = att + (size_t)NATM * HID;     // 384*128
  float* O   = (float*)d_out;

  hipLaunchKernelGGL(hp_gemm, dim3(HPC / 16, NATM / 16), dim3(32), 0, stream,
                     h, Wfc, hp);
  hipLaunchKernelGGL(pair_reduce, dim3(NATM), dim3(256), 0, stream,
                     x, hp, att);
  hipLaunchKernelGGL(out_gemm, dim3(OUTC / 16, NATM / 16), dim3(32), 0, stream,
                     att, Wsum, bsum, out);
  const unsigned total = (unsigned)NATM * NATM * RBF;
  hipLaunchKernelGGL(emit, dim3((total + 255) / 256), dim3(256), 0, stream,
                     out, O);
}